// GaussianSplat3D_61211873903267
// MI455X (gfx1250) — compile-verified
//
#include <hip/hip_runtime.h>
#include <stdint.h>

// ---------------------------------------------------------------------------
// Gaussian splatting forward renderer for MI455X (gfx1250).
// Phase 3 stages the whole sorted per-gaussian parameter table (48 KB) into
// LDS with a single Tensor Data Mover DMA per workgroup, then composites.
// ---------------------------------------------------------------------------

typedef unsigned int u32x4 __attribute__((ext_vector_type(4)));
typedef int          i32x8 __attribute__((ext_vector_type(8)));
typedef int          i32x4 __attribute__((ext_vector_type(4)));

#define GS_NMAX   1024
#define GS_STRIDE 12           // floats per packed gaussian (9 used + pad), 48B
#define IMG_W     224
#define IMG_H     224

// ---------------- Phase 1: per-gaussian preprocessing ----------------------
__global__ __launch_bounds__(256) void gs_preprocess(
    const float* __restrict__ means, const float* __restrict__ log_scales,
    const float* __restrict__ quats, const float* __restrict__ opac_logits,
    const float* __restrict__ sh,    const float* __restrict__ viewmats,
    const float* __restrict__ Ks,    int n,
    float* __restrict__ packed,      float* __restrict__ depth)
{
  int i = blockIdx.x * blockDim.x + threadIdx.x;
  if (i >= n) return;

  const float nearp = 0.01f, farp = 1e10f, eps2d = 0.3f;

  float Rm[3][3] = {
    {viewmats[0], viewmats[1], viewmats[2]},
    {viewmats[4], viewmats[5], viewmats[6]},
    {viewmats[8], viewmats[9], viewmats[10]}};
  float t0 = viewmats[3], t1 = viewmats[7], t2 = viewmats[11];
  float fx = Ks[0], cx = Ks[2], fy = Ks[4], cy = Ks[5];

  float mx = means[3*i], my = means[3*i+1], mz = means[3*i+2];
  float x = Rm[0][0]*mx + Rm[0][1]*my + Rm[0][2]*mz + t0;
  float y = Rm[1][0]*mx + Rm[1][1]*my + Rm[1][2]*mz + t1;
  float z = Rm[2][0]*mx + Rm[2][1]*my + Rm[2][2]*mz + t2;
  float zs = (z > nearp) ? z : 1.0f;
  float m2x = fx * x / zs + cx;
  float m2y = fy * y / zs + cy;

  // quaternion (w,x,y,z) -> rotation
  float qw = quats[4*i], qx = quats[4*i+1], qy = quats[4*i+2], qz = quats[4*i+3];
  float qinv = 1.0f / sqrtf(qw*qw + qx*qx + qy*qy + qz*qz);
  qw *= qinv; qx *= qinv; qy *= qinv; qz *= qinv;
  float Rg[3][3] = {
    {1.f - 2.f*(qy*qy + qz*qz), 2.f*(qx*qy - qw*qz),       2.f*(qx*qz + qw*qy)},
    {2.f*(qx*qy + qw*qz),       1.f - 2.f*(qx*qx + qz*qz), 2.f*(qy*qz - qw*qx)},
    {2.f*(qx*qz - qw*qy),       2.f*(qy*qz + qw*qx),       1.f - 2.f*(qx*qx + qy*qy)}};

  float s0 = __expf(log_scales[3*i]);
  float s1 = __expf(log_scales[3*i+1]);
  float s2 = __expf(log_scales[3*i+2]);

  float M[3][3];
  #pragma unroll
  for (int a = 0; a < 3; ++a) {
    M[a][0] = Rg[a][0] * s0; M[a][1] = Rg[a][1] * s1; M[a][2] = Rg[a][2] * s2;
  }
  float cov3[3][3];
  #pragma unroll
  for (int a = 0; a < 3; ++a)
    #pragma unroll
    for (int b = 0; b < 3; ++b)
      cov3[a][b] = M[a][0]*M[b][0] + M[a][1]*M[b][1] + M[a][2]*M[b][2];

  // covc = R * cov3 * R^T
  float tmp[3][3], covc[3][3];
  #pragma unroll
  for (int a = 0; a < 3; ++a)
    #pragma unroll
    for (int b = 0; b < 3; ++b)
      tmp[a][b] = Rm[a][0]*cov3[0][b] + Rm[a][1]*cov3[1][b] + Rm[a][2]*cov3[2][b];
  #pragma unroll
  for (int a = 0; a < 3; ++a)
    #pragma unroll
    for (int b = 0; b < 3; ++b)
      covc[a][b] = tmp[a][0]*Rm[b][0] + tmp[a][1]*Rm[b][1] + tmp[a][2]*Rm[b][2];

  float limx = 1.3f * 0.5f * (float)IMG_W / fx;
  float limy = 1.3f * 0.5f * (float)IMG_H / fy;
  float tx = zs * fminf(fmaxf(x / zs, -limx), limx);
  float ty = zs * fminf(fmaxf(y / zs, -limy), limy);
  float rz = 1.f / zs, rz2 = rz * rz;
  float j00 = fx * rz, j02 = -fx * tx * rz2;
  float j11 = fy * rz, j12 = -fy * ty * rz2;

  float jc00 = j00*covc[0][0] + j02*covc[2][0];
  float jc01 = j00*covc[0][1] + j02*covc[2][1];
  float jc02 = j00*covc[0][2] + j02*covc[2][2];
  float jc10 = j11*covc[1][0] + j12*covc[2][0];
  float jc11 = j11*covc[1][1] + j12*covc[2][1];
  float jc12 = j11*covc[1][2] + j12*covc[2][2];

  float a = jc00*j00 + jc02*j02 + eps2d;            // cov2[0][0]+eps
  float b = jc01*j11 + jc02*j12;                    // cov2[0][1]
  float c = jc11*j11 + jc12*j12 + eps2d;            // cov2[1][1]+eps
  float det = a*c - b*b;
  float det_s = (det > 0.f) ? det : 1.f;
  float inv = 1.f / det_s;
  float cA = c * inv, cB = -b * inv, cC = a * inv;
  float mid = 0.5f * (a + c);
  float radius = ceilf(3.f * sqrtf(mid + sqrtf(fmaxf(mid*mid - det, 0.01f))));

  bool inside = (m2x + radius > 0.f) && (m2x - radius < (float)IMG_W) &&
                (m2y + radius > 0.f) && (m2y - radius < (float)IMG_H);
  bool visible = (z > nearp) && (z < farp) && (det > 0.f) && (radius > 0.f) && inside;

  // camera position = -R^T t ; view direction
  float cpx = -(Rm[0][0]*t0 + Rm[1][0]*t1 + Rm[2][0]*t2);
  float cpy = -(Rm[0][1]*t0 + Rm[1][1]*t1 + Rm[2][1]*t2);
  float cpz = -(Rm[0][2]*t0 + Rm[1][2]*t1 + Rm[2][2]*t2);
  float dX = mx - cpx, dY = my - cpy, dZ = mz - cpz;
  float dn = 1.f / sqrtf(dX*dX + dY*dY + dZ*dZ);
  dX *= dn; dY *= dn; dZ *= dn;

  // degree-3 SH basis
  const float C0 = 0.28209479177387814f;
  const float C1 = 0.4886025119029199f;
  const float C2_0 = 1.0925484305920792f, C2_1 = -1.0925484305920792f,
              C2_2 = 0.31539156525252005f, C2_3 = -1.0925484305920792f,
              C2_4 = 0.5462742152960396f;
  const float C3_0 = -0.5900435899266435f, C3_1 = 2.890611442640554f,
              C3_2 = -0.4570457994644658f, C3_3 = 0.3731763325901154f,
              C3_4 = -0.4570457994644658f, C3_5 = 1.445305721320277f,
              C3_6 = -0.5900435899266435f;
  float xx = dX*dX, yy = dY*dY, zz = dZ*dZ;
  float xy = dX*dY, yzp = dY*dZ, xzp = dX*dZ;
  float basis[16];
  basis[0]  = C0;
  basis[1]  = -C1 * dY;
  basis[2]  =  C1 * dZ;
  basis[3]  = -C1 * dX;
  basis[4]  = C2_0 * xy;
  basis[5]  = C2_1 * yzp;
  basis[6]  = C2_2 * (2.f*zz - xx - yy);
  basis[7]  = C2_3 * xzp;
  basis[8]  = C2_4 * (xx - yy);
  basis[9]  = C3_0 * (dY * (3.f*xx - yy));
  basis[10] = C3_1 * (xy * dZ);
  basis[11] = C3_2 * (dY * (4.f*zz - xx - yy));
  basis[12] = C3_3 * (dZ * (2.f*zz - 3.f*xx - 3.f*yy));
  basis[13] = C3_4 * (dX * (4.f*zz - xx - yy));
  basis[14] = C3_5 * (dZ * (xx - yy));
  basis[15] = C3_6 * (dX * (xx - yy));

  float col[3];
  #pragma unroll
  for (int ch = 0; ch < 3; ++ch) {
    float acc = 0.f;
    #pragma unroll
    for (int k = 0; k < 16; ++k)
      acc = fmaf(basis[k], sh[(k * n + i) * 3 + ch], acc);
    col[ch] = fmaxf(acc + 0.5f, 0.f);
  }

  float op = visible ? (1.f / (1.f + __expf(-opac_logits[i]))) : 0.f;

  float* o = packed + i * GS_STRIDE;
  o[0] = m2x; o[1] = m2y; o[2] = cA; o[3] = cB; o[4] = cC;
  o[5] = op;  o[6] = col[0]; o[7] = col[1]; o[8] = col[2];
  o[9] = 0.f; o[10] = 0.f; o[11] = 0.f;
  depth[i] = visible ? z : __builtin_inff();
}

// ---------------- Phase 2: bitonic depth sort + gather ---------------------
__global__ __launch_bounds__(GS_NMAX) void gs_sort_gather(
    const float* __restrict__ depth, const float* __restrict__ packed,
    float* __restrict__ sorted, int n)
{
  __shared__ float skey[GS_NMAX];
  __shared__ int   sidx[GS_NMAX];
  int tid = threadIdx.x;
  skey[tid] = (tid < n) ? depth[tid] : __builtin_inff();
  sidx[tid] = tid;
  __syncthreads();

  for (int k = 2; k <= GS_NMAX; k <<= 1) {
    for (int j = k >> 1; j > 0; j >>= 1) {
      int ixj = tid ^ j;
      if (ixj > tid) {
        bool up = ((tid & k) == 0);
        float ka = skey[tid], kb = skey[ixj];
        int ia = sidx[tid], ib = sidx[ixj];
        // ascending with index tiebreak (stable-argsort equivalent)
        bool greater = (ka > kb) || (ka == kb && ia > ib);
        if (up == greater) {
          skey[tid] = kb; skey[ixj] = ka;
          sidx[tid] = ib; sidx[ixj] = ia;
        }
      }
      __syncthreads();
    }
  }

  int src = sidx[tid];
  if (src < n) {
    #pragma unroll
    for (int kk = 0; kk < GS_STRIDE; ++kk)
      sorted[tid * GS_STRIDE + kk] = packed[src * GS_STRIDE + kk];
  } else {
    #pragma unroll
    for (int kk = 0; kk < GS_STRIDE; ++kk)
      sorted[tid * GS_STRIDE + kk] = 0.f;   // opac==0 -> no contribution
  }
}

// ---------------- Phase 3: per-pixel compositing (TDM-staged LDS) ----------
__global__ __launch_bounds__(256) void gs_composite(
    const float* __restrict__ sorted, float* __restrict__ out)
{
  __shared__ float sg[GS_NMAX * GS_STRIDE];   // 48 KB

  // One TDM DMA per workgroup: pull the whole sorted parameter table into
  // LDS as a 1D tile of 12288 dwords. Issued by wave 0 only (TDM ignores
  // EXEC; TENSORcnt is per-wave so the issuing wave waits, then barrier).
  if (threadIdx.x < 32) {
    uint64_t ga  = (uint64_t)(uintptr_t)sorted;
    unsigned lds = (unsigned)(uintptr_t)&sg[0];     // low 32 bits = LDS offset
    const unsigned nd = GS_NMAX * GS_STRIDE;        // 12288 dwords

    u32x4 g0;
    g0.x = 1u;                                      // count=1 valid descriptor
    g0.y = lds;                                     // lds_addr [63:32]
    g0.z = (unsigned)(ga & 0xFFFFFFFFu);            // global_addr [95:64]
    g0.w = ((unsigned)(ga >> 32) & 0x01FFFFFFu)     // global_addr [120:96]
         | 0x80000000u;                             // type=2 ("image") [127:126]

    i32x8 g1;
    g1.s0 = (int)(2u << 16);                        // data_size=2 -> 4 bytes
    g1.s1 = (int)((nd & 0xFFFFu) << 16);            // tensor_dim0 lo16
    g1.s2 = (int)((nd >> 16) | (1u << 16));         // tensor_dim0 hi16 | tensor_dim1=1
    g1.s3 = (int)((nd & 0xFFFFu) << 16);            // tile_dim0 = nd
    g1.s4 = 1;                                      // tile_dim1=1, tile_dim2=0
    g1.s5 = (int)nd;                                // tensor_dim0_stride lo32
    g1.s6 = (int)((nd & 0xFFFFu) << 16);            // stride0 hi16=0 | stride1 lo16
    g1.s7 = 0;                                      // stride1 hi32
    i32x4 gz4 = {0, 0, 0, 0};                       // groups 2/3 unused (<=2D)
    i32x8 gz8 = {0, 0, 0, 0, 0, 0, 0, 0};           // extra group (6-arg form)

    __builtin_amdgcn_tensor_load_to_lds(g0, g1, gz4, gz4, gz8, 0);
    __builtin_amdgcn_s_wait_tensorcnt(0);
  }
  __syncthreads();

  int p = blockIdx.x * blockDim.x + threadIdx.x;
  if (p >= IMG_W * IMG_H) return;
  int pyi = p / IMG_W, pxi = p - pyi * IMG_W;
  float px = (float)pxi + 0.5f, py = (float)pyi + 0.5f;

  float T = 1.f, rr = 0.f, gg = 0.f, bb = 0.f;
  // Unroll x4: consecutive gaussians' sigma/exp are independent (only the
  // 1-FMA T-chain is serial), letting VALU work of gaussian g+1 hide the
  // v_exp_f32 TRANS latency of gaussian g.
  #pragma unroll 4
  for (int gi = 0; gi < GS_NMAX; ++gi) {
    // 48B record, 16B aligned: two ds_load_b128 + one ds_load_b32 broadcast
    const float4* q = reinterpret_cast<const float4*>(&sg[gi * GS_STRIDE]);
    float4 e0 = q[0];                     // m2x, m2y, cA, cB
    float4 e1 = q[1];                     // cC, opac, colR, colG
    float  cb = sg[gi * GS_STRIDE + 8];   // colB
    float dx = px - e0.x;
    float dy = py - e0.y;
    float sigma = 0.5f * (e0.z*dx*dx + e1.x*dy*dy) + e0.w*dx*dy;
    float al = e1.y * __expf(-sigma);
    if (sigma < 0.f || al < (1.f / 255.f)) al = 0.f;
    else                                   al = fminf(al, 0.999f);
    float w = al * T;
    rr = fmaf(w, e1.z, rr);
    gg = fmaf(w, e1.w, gg);
    bb = fmaf(w, cb,  bb);
    T *= (1.f - al);
  }

  out[3*p + 0] = rr;
  out[3*p + 1] = gg;
  out[3*p + 2] = bb;
  out[IMG_W * IMG_H * 3 + p] = 1.f - T;             // alpha image
}

// ---------------------------------------------------------------------------
extern "C" void kernel_launch(void* const* d_in, const int* in_sizes, int n_in,
                              void* d_out, int out_size, void* d_ws, size_t ws_size,
                              hipStream_t stream) {
  const float* means       = (const float*)d_in[0];
  const float* log_scales  = (const float*)d_in[1];
  const float* quats       = (const float*)d_in[2];
  const float* opac_logits = (const float*)d_in[3];
  const float* sh          = (const float*)d_in[4];
  const float* viewmats    = (const float*)d_in[5];
  const float* Ks          = (const float*)d_in[6];
  int n = in_sizes[0] / 3;
  if (n > GS_NMAX) n = GS_NMAX;

  float* packed = (float*)d_ws;                       // 1024*12 floats
  float* depth  = packed + GS_NMAX * GS_STRIDE;       // 1024 floats
  float* sorted = depth + GS_NMAX;                    // 1024*12 floats

  gs_preprocess<<<(n + 255) / 256, 256, 0, stream>>>(
      means, log_scales, quats, opac_logits, sh, viewmats, Ks, n, packed, depth);
  gs_sort_gather<<<1, GS_NMAX, 0, stream>>>(depth, packed, sorted, n);
  gs_composite<<<(IMG_W * IMG_H + 255) / 256, 256, 0, stream>>>(sorted, (float*)d_out);
}